// Encoder_81561428951638
// MI455X (gfx1250) — compile-verified
//
#include <hip/hip_runtime.h>

typedef __attribute__((ext_vector_type(2))) float v2f;
typedef __attribute__((ext_vector_type(8))) float v8f;

#define HD 128
#define WT_STRIDE 132   // 132 mod 64 = 4 -> conflict-free ds_load_b64 B-fragments

// ---------------- degree / normalization ----------------
__global__ __launch_bounds__(256) void k_deg_init(float* __restrict__ deg, int n) {
  int i = blockIdx.x * 256 + threadIdx.x;
  if (i < n) deg[i] = 1.0f;  // self-loop
}

__global__ __launch_bounds__(256) void k_deg_count(const int* __restrict__ dst,
                                                   float* __restrict__ deg, int e) {
  int stride = gridDim.x * 256;
  for (int i = blockIdx.x * 256 + threadIdx.x; i < e; i += stride)
    atomicAdd(&deg[dst[i]], 1.0f);
}

__global__ __launch_bounds__(256) void k_dinv(float* __restrict__ deg, int n) {
  int i = blockIdx.x * 256 + threadIdx.x;
  if (i < n) deg[i] = rsqrtf(deg[i]);  // deg >= 1 always
}

// ---------------- fp32 WMMA GEMM: out[n,128] = A[n,128] @ W[128,128] ----------
__global__ __launch_bounds__(256) void k_gemm_wmma(const float* __restrict__ A,
                                                   const float* __restrict__ W,
                                                   float* __restrict__ out, int n) {
  __shared__ float Wt[HD * WT_STRIDE];  // W transposed: Wt[col][k]
  for (int i = threadIdx.x; i < HD * HD / 4; i += 256) {
    int k  = i >> 5;
    int n4 = (i & 31) * 4;
    const float4 w = *(const float4*)(W + k * HD + n4);
    Wt[(n4 + 0) * WT_STRIDE + k] = w.x;
    Wt[(n4 + 1) * WT_STRIDE + k] = w.y;
    Wt[(n4 + 2) * WT_STRIDE + k] = w.z;
    Wt[(n4 + 3) * WT_STRIDE + k] = w.w;
  }
  __syncthreads();

  const int lane = threadIdx.x & 31;
  const int wave = threadIdx.x >> 5;
  const int r0   = blockIdx.x * 128 + wave * 16;   // 16 rows per wave
  if (r0 + 16 > n) return;                          // wave-uniform (EXEC stays all-1)
  const int hi = lane >> 4;                         // K half-select
  const int lo = lane & 15;
  const float* arow = A + (size_t)(r0 + lo) * HD;

  v8f acc[8] = {};                                  // 8 N-tiles of 16x16 f32
  for (int kk = 0; kk < 32; ++kk) {
    const int k = kk * 4 + hi * 2;
    // A 16x4 frag: lanes 0-15 hold K={k,k+1} of row lo; lanes 16-31 K={k+2,k+3}
    const v2f a = *(const v2f*)(arow + k);
#pragma unroll
    for (int t = 0; t < 8; ++t) {
      // B 4x16 frag: lanes 0-15 -> W[k..k+1][col], lanes 16-31 -> W[k+2..k+3][col]
      const v2f b = *(const v2f*)(&Wt[(t * 16 + lo) * WT_STRIDE + k]);
      acc[t] = __builtin_amdgcn_wmma_f32_16x16x4_f32(
          false, a, false, b, (short)0, acc[t], false, false);
    }
  }
#pragma unroll
  for (int t = 0; t < 8; ++t) {
    float* orow = out + (size_t)(r0 + hi * 8) * HD + t * 16 + lo;
#pragma unroll
    for (int r = 0; r < 8; ++r)
      orow[(size_t)r * HD] = acc[t][r];  // C/D layout: VGPR r -> rows r / r+8
  }
}

// ---------------- self-loop init / bias / relu ----------------
__global__ __launch_bounds__(256) void k_self_init(const float* __restrict__ dinv,
                                                   const float* __restrict__ m,
                                                   float* __restrict__ agg, int total) {
  int i = blockIdx.x * 256 + threadIdx.x;
  if (i < total) {
    float di = dinv[i >> 7];
    agg[i] = di * di * m[i];
  }
}

__global__ __launch_bounds__(256) void k_self_init_bias(const float* __restrict__ dinv,
                                                        const float* __restrict__ m,
                                                        const float* __restrict__ bias,
                                                        float* __restrict__ outp, int total) {
  int i = blockIdx.x * 256 + threadIdx.x;
  if (i < total) {
    float di = dinv[i >> 7];
    outp[i] = di * di * m[i] + bias[i & 127];
  }
}

__global__ __launch_bounds__(256) void k_bias_relu(float* __restrict__ agg,
                                                   const float* __restrict__ bias, int total) {
  int i = blockIdx.x * 256 + threadIdx.x;
  if (i < total) {
    float v = agg[i] + bias[i & 127];
    agg[i] = v > 0.0f ? v : 0.0f;
  }
}

// ---------------- edge scatter: agg[dst] += dinv[s]*dinv[d]*m[src] ------------
__global__ __launch_bounds__(256) void k_scatter(const int* __restrict__ src,
                                                 const int* __restrict__ dst,
                                                 const float* __restrict__ dinv,
                                                 const float* __restrict__ m,
                                                 float* __restrict__ agg, int e) {
  const int lane = threadIdx.x & 31;               // wave32: lane covers 4 floats
  int w = (blockIdx.x * 256 + threadIdx.x) >> 5;
  const int nw = (gridDim.x * 256) >> 5;
  for (; w < e; w += nw) {
    const int s = src[w];
    const int d = dst[w];
    const float nrm = dinv[s] * dinv[d];
    const float4 v = *(const float4*)(m + (size_t)s * HD + lane * 4);
    float* p = agg + (size_t)d * HD + lane * 4;
    atomicAdd(p + 0, nrm * v.x);   // f32 atomics resolve in 192MB L2 (51MB buffer)
    atomicAdd(p + 1, nrm * v.y);
    atomicAdd(p + 2, nrm * v.z);
    atomicAdd(p + 3, nrm * v.w);
  }
}

// ---------------- launcher ----------------
extern "C" void kernel_launch(void* const* d_in, const int* in_sizes, int n_in,
                              void* d_out, int out_size, void* d_ws, size_t ws_size,
                              hipStream_t stream) {
  const float* x  = (const float*)d_in[0];
  const float* W1 = (const float*)d_in[1];
  const float* b1 = (const float*)d_in[2];
  const float* W2 = (const float*)d_in[3];
  const float* b2 = (const float*)d_in[4];
  const int*   ei = (const int*)d_in[5];

  const int n = in_sizes[0] / HD;        // 100000 (divisible by 16)
  const int e = in_sizes[5] / 2;         // 1600000
  const int* src = ei;
  const int* dst = ei + e;
  const int total = n * HD;

  char* ws = (char*)d_ws;
  float* dinv = (float*)ws;                                        // n floats
  size_t off1 = ((size_t)n * 4 + 255) & ~(size_t)255;
  float* buf1 = (float*)(ws + off1);                               // n*128 (GEMM out)
  size_t off2 = off1 + (((size_t)total * 4 + 255) & ~(size_t)255);
  float* buf2 = (float*)(ws + off2);                               // n*128 (agg / h)
  float* out  = (float*)d_out;

  dim3 blk(256);
  const int gN   = (n + 255) / 256;
  const int gNH  = (total + 255) / 256;
  const int gMM  = (n + 127) / 128;

  // normalization
  k_deg_init <<<gN,   blk, 0, stream>>>(dinv, n);
  k_deg_count<<<2048, blk, 0, stream>>>(dst, dinv, e);
  k_dinv     <<<gN,   blk, 0, stream>>>(dinv, n);

  // layer 1: m = x@W1 ; agg = D^-1/2 A_hat D^-1/2 m ; h = relu(agg + b1)
  k_gemm_wmma<<<gMM,  blk, 0, stream>>>(x, W1, buf1, n);
  k_self_init<<<gNH,  blk, 0, stream>>>(dinv, buf1, buf2, total);
  k_scatter  <<<4096, blk, 0, stream>>>(src, dst, dinv, buf1, buf2, e);
  k_bias_relu<<<gNH,  blk, 0, stream>>>(buf2, b1, total);

  // layer 2: m2 = h@W2 ; out = D^-1/2 A_hat D^-1/2 m2 + b2
  k_gemm_wmma<<<gMM,  blk, 0, stream>>>(buf2, W2, buf1, n);
  k_self_init_bias<<<gNH, blk, 0, stream>>>(dinv, buf1, b2, out, total);
  k_scatter  <<<4096, blk, 0, stream>>>(src, dst, dinv, buf1, out, e);
}